// MODEL_31628139168141
// MI455X (gfx1250) — compile-verified
//
#include <hip/hip_runtime.h>

#define NNODES 100000
#define BATCH  2048
#define DTOT   240
#define NEG_SLOPE 0.01f

typedef float v2f __attribute__((ext_vector_type(2)));
typedef float v8f __attribute__((ext_vector_type(8)));

__device__ __forceinline__ float leaky(float x) {
    return x >= 0.0f ? x : NEG_SLOPE * x;
}

// Native CDNA5 fp32 scatter-add (no-return form, STOREcnt-tracked).
__device__ __forceinline__ void atomic_fadd_f32(float* addr, float val) {
    asm volatile("global_atomic_add_f32 %0, %1, off"
                 :
                 : "v"(addr), "v"(val)
                 : "memory");
}

// ---------------- utility kernels ----------------

__global__ void k_zero(float* p, long long n) {
    long long i = (long long)blockIdx.x * blockDim.x + threadIdx.x;
    if (i < n) p[i] = 0.0f;
}

// copy emb [N,128] into all_embed [N,240] columns 0..127
__global__ void k_copy_emb(const float* __restrict__ emb, float* __restrict__ all_embed) {
    long long i = (long long)blockIdx.x * blockDim.x + threadIdx.x;
    if (i >= (long long)NNODES * 128) return;
    int row = (int)(i >> 7);
    int col = (int)(i & 127);
    all_embed[(long long)row * DTOT + col] = emb[i];
}

// ---------------- SpMM (COO scatter-add) ----------------
// side[row, :] += A_vals[e] * ego[col, :]
// one thread per (edge, column-pair): float2 gather + 2 native fp32 atomics
__global__ void k_spmm(const int* __restrict__ row_idx, const int* __restrict__ col_idx,
                       const float* __restrict__ vals, const float* __restrict__ ego,
                       float* __restrict__ side, int E, int pair_log2) {
    // pair_log2 = log2(din/2); din = 2 << pair_log2
    int npair = 1 << pair_log2;
    int t = blockIdx.x * blockDim.x + threadIdx.x;
    int e = t >> pair_log2;
    int jp = t & (npair - 1);          // column pair index
    if (e >= E) return;
    int r = row_idx[e];
    int c = col_idx[e];
    float a = vals[e];
    long long din = (long long)npair * 2;
    const v2f* src = (const v2f*)(ego + (long long)c * din) + jp;
    v2f v = *src;
    float* dst = side + (long long)r * din + (long long)jp * 2;
    atomic_fadd_f32(dst + 0, a * v.x);
    atomic_fadd_f32(dst + 1, a * v.y);
}

// ---------------- layer GEMM with fp32 WMMA ----------------
// out[m,n] = leaky( sum_k (ego+side)[m,k]*W1[n,k] + b1[n] )
//          + leaky( sum_k (ego*side)[m,k]*W2[n,k] + b2[n] )
// one wave32 per 16x16 output tile; V_WMMA_F32_16X16X4_F32
__global__ void k_layer_gemm(const float* __restrict__ ego, const float* __restrict__ side,
                             const float* __restrict__ W1, const float* __restrict__ b1,
                             const float* __restrict__ W2, const float* __restrict__ b2,
                             float* __restrict__ out, int din, int dout, int colTiles) {
    int tile = blockIdx.x;
    int rowTile = tile / colTiles;
    int colTile = tile - rowTile * colTiles;
    int lane = threadIdx.x;
    int m    = lane & 15;   // A rows / B,C columns striped across lanes
    int half = lane >> 4;   // A/B: lanes 16-31 hold K = k0+2,k0+3

    int row  = rowTile * 16 + m;   // node index for A-frag load
    int ncol = colTile * 16 + m;   // output column for B-frag load

    const float* erow  = ego  + (long long)row  * din;
    const float* srow  = side + (long long)row  * din;
    const float* w1row = W1   + (long long)ncol * din;
    const float* w2row = W2   + (long long)ncol * din;

    v8f c1 = {0.f,0.f,0.f,0.f,0.f,0.f,0.f,0.f};
    v8f c2 = {0.f,0.f,0.f,0.f,0.f,0.f,0.f,0.f};

    for (int k0 = 0; k0 < din; k0 += 4) {
        int ka = k0 + half * 2;
        float ex = erow[ka], ey = erow[ka + 1];
        float sx = srow[ka], sy = srow[ka + 1];
        v2f a1 = {ex + sx, ey + sy};   // sum path A-frag
        v2f a2 = {ex * sx, ey * sy};   // bi  path A-frag
        v2f bb1 = {w1row[ka], w1row[ka + 1]};   // B[k][n] = W[n][k]
        v2f bb2 = {w2row[ka], w2row[ka + 1]};
        c1 = __builtin_amdgcn_wmma_f32_16x16x4_f32(false, a1, false, bb1, (short)0, c1, false, false);
        c2 = __builtin_amdgcn_wmma_f32_16x16x4_f32(false, a2, false, bb2, (short)0, c2, false, false);
    }

    float bias1 = b1[ncol];
    float bias2 = b2[ncol];
    // C/D layout: VGPR r -> M = r (lanes 0-15) / r+8 (lanes 16-31); N = lane&15
    for (int r = 0; r < 8; ++r) {
        int orow = rowTile * 16 + r + 8 * half;
        float v = leaky(c1[r] + bias1) + leaky(c2[r] + bias2);
        out[(long long)orow * dout + ncol] = v;
    }
}

// ---------------- row L2-normalize into all_embed slice ----------------
__global__ void k_norm(const float* __restrict__ ego, float* __restrict__ all_embed,
                       int dout, int coloff) {
    int gid  = blockIdx.x * blockDim.x + threadIdx.x;
    int wave = gid >> 5;
    int lane = gid & 31;
    if (wave >= NNODES) return;
    const float* row = ego + (long long)wave * dout;
    float s = 0.f;
    for (int j = lane; j < dout; j += 32) { float v = row[j]; s += v * v; }
    for (int off = 16; off; off >>= 1) s += __shfl_xor(s, off, 32);
    float inv = 1.0f / fmaxf(sqrtf(s), 1e-12f);
    float* outrow = all_embed + (long long)wave * DTOT + coloff;
    for (int j = lane; j < dout; j += 32) outrow[j] = row[j] * inv;
}

// ---------------- gather batch rows ----------------
__global__ void k_gather(const int* __restrict__ ids, const float* __restrict__ all_embed,
                         float* __restrict__ dst) {
    int i = blockIdx.x * blockDim.x + threadIdx.x;
    if (i >= BATCH * DTOT) return;
    int b = i / DTOT;
    int c = i - b * DTOT;
    dst[i] = all_embed[(long long)ids[b] * DTOT + c];
}

// ---------------- final score GEMM: out = P @ D^T ----------------
__global__ void k_score_gemm(const float* __restrict__ P, const float* __restrict__ Dm,
                             float* __restrict__ out) {
    int tile = blockIdx.x;            // 128 x 128 tiles of 16x16
    int rowTile = tile >> 7;
    int colTile = tile & 127;
    int lane = threadIdx.x;
    int m    = lane & 15;
    int half = lane >> 4;

    const float* prow = P  + (long long)(rowTile * 16 + m) * DTOT;
    const float* drow = Dm + (long long)(colTile * 16 + m) * DTOT; // B[k][n] = D[n][k]

    v8f acc = {0.f,0.f,0.f,0.f,0.f,0.f,0.f,0.f};
    for (int k0 = 0; k0 < DTOT; k0 += 4) {
        int ka = k0 + half * 2;
        v2f a = {prow[ka], prow[ka + 1]};
        v2f b = {drow[ka], drow[ka + 1]};
        acc = __builtin_amdgcn_wmma_f32_16x16x4_f32(false, a, false, b, (short)0, acc, false, false);
    }
    int ncol = colTile * 16 + m;
    for (int r = 0; r < 8; ++r) {
        int orow = rowTile * 16 + r + 8 * half;
        out[(long long)orow * BATCH + ncol] = acc[r];
    }
}

// ---------------- host launcher ----------------
extern "C" void kernel_launch(void* const* d_in, const int* in_sizes, int n_in,
                              void* d_out, int out_size, void* d_ws, size_t ws_size,
                              hipStream_t stream) {
    const int*   patient_ids = (const int*)d_in[0];
    const int*   disease_ids = (const int*)d_in[1];
    const int*   row_idx     = (const int*)d_in[2];
    const int*   col_idx     = (const int*)d_in[3];
    const float* A_vals      = (const float*)d_in[4];
    const float* emb         = (const float*)d_in[5];
    const float* W1[3] = {(const float*)d_in[6],  (const float*)d_in[10], (const float*)d_in[14]};
    const float* b1[3] = {(const float*)d_in[7],  (const float*)d_in[11], (const float*)d_in[15]};
    const float* W2[3] = {(const float*)d_in[8],  (const float*)d_in[12], (const float*)d_in[16]};
    const float* b2[3] = {(const float*)d_in[9],  (const float*)d_in[13], (const float*)d_in[17]};
    int E = in_sizes[2];

    float* ws        = (float*)d_ws;
    float* all_embed = ws;                                   // N*240
    float* ego_a     = all_embed + (size_t)NNODES * DTOT;    // N*64
    float* ego_b     = ego_a + (size_t)NNODES * 64;          // N*32
    float* side      = ego_b + (size_t)NNODES * 32;          // N*128
    float* p_buf     = side + (size_t)NNODES * 128;          // 2048*240
    float* d_buf     = p_buf + (size_t)BATCH * DTOT;

    // all_embed[:, 0:128] = emb
    {
        long long n = (long long)NNODES * 128;
        k_copy_emb<<<(unsigned)((n + 255) / 256), 256, 0, stream>>>(emb, all_embed);
    }

    const int dins[3]    = {128, 64, 32};
    const int plog[3]    = {6, 5, 4};    // log2(din/2)
    const int douts[3]   = {64, 32, 16};
    const int coloffs[3] = {128, 192, 224};
    const float* ego_in = emb;
    float* ego_out[3] = {ego_a, ego_b, ego_a};

    for (int l = 0; l < 3; ++l) {
        int din = dins[l], dout = douts[l];
        long long nside = (long long)NNODES * din;
        k_zero<<<(unsigned)((nside + 255) / 256), 256, 0, stream>>>(side, nside);

        long long tot = (long long)E << plog[l];   // E * din/2 threads
        k_spmm<<<(unsigned)((tot + 255) / 256), 256, 0, stream>>>(
            row_idx, col_idx, A_vals, ego_in, side, E, plog[l]);

        int colTiles = dout / 16;
        int rowTiles = NNODES / 16;   // 6250 exact
        k_layer_gemm<<<rowTiles * colTiles, 32, 0, stream>>>(
            ego_in, side, W1[l], b1[l], W2[l], b2[l], ego_out[l], din, dout, colTiles);

        long long nthr = (long long)NNODES * 32;
        k_norm<<<(unsigned)((nthr + 255) / 256), 256, 0, stream>>>(
            ego_out[l], all_embed, dout, coloffs[l]);

        ego_in = ego_out[l];
    }

    int ng = (BATCH * DTOT + 255) / 256;
    k_gather<<<ng, 256, 0, stream>>>(patient_ids, all_embed, p_buf);
    k_gather<<<ng, 256, 0, stream>>>(disease_ids, all_embed, d_buf);

    k_score_gemm<<<128 * 128, 32, 0, stream>>>(p_buf, d_buf, (float*)d_out);
}